// TaskAttention_57260503990882
// MI455X (gfx1250) — compile-verified
//
#include <hip/hip_runtime.h>

typedef float v2f __attribute__((ext_vector_type(2)));
typedef float v4f __attribute__((ext_vector_type(4)));
typedef float v8f __attribute__((ext_vector_type(8)));

#define B_ 64
#define N_ 1024
#define D_ 256
#define T_ 4

// ---------------------------------------------------------------------------
// K1: qk2[t][s][d] = sum_o (Wq[t]@emb[s] + bq[t])[o] * Wk[t][o][d]
// 16 blocks (t*4+s), 256 threads. Trivial cost (~2M MACs).
// ---------------------------------------------------------------------------
__global__ void __launch_bounds__(256)
k_precompute_qk(const float* __restrict__ emb,
                const float* __restrict__ Wq,
                const float* __restrict__ bq,
                const float* __restrict__ Wk,
                float* __restrict__ qk2) {
    __shared__ float e_lds[D_];
    __shared__ float q_lds[D_];
    const int t = blockIdx.x >> 2;
    const int s = blockIdx.x & 3;
    const int tid = threadIdx.x;

    e_lds[tid] = emb[s * D_ + tid];
    __syncthreads();

    // q[o] = bq[t][o] + dot(emb[s], Wq[t][o][:])
    const float* wrow = Wq + ((size_t)t * D_ + tid) * D_;
    float acc = bq[t * D_ + tid];
    for (int d = 0; d < D_; d += 4) {
        v4f w = *(const v4f*)(wrow + d);
        acc += w.x * e_lds[d] + w.y * e_lds[d + 1] +
               w.z * e_lds[d + 2] + w.w * e_lds[d + 3];
    }
    q_lds[tid] = acc;
    __syncthreads();

    // qk2[d] = sum_o q[o] * Wk[t][o][d]   (coalesced over d)
    float a2 = 0.f;
    const float* wk = Wk + (size_t)t * D_ * D_ + tid;
    for (int o = 0; o < D_; ++o) a2 += q_lds[o] * wk[(size_t)o * D_];
    qk2[(size_t)blockIdx.x * D_ + tid] = a2;
}

// ---------------------------------------------------------------------------
// K2: one block per batch b. WMMA f32 16x16x4 scores GEMM (M=1024,N=4pad16,
// K=256), 4 softmaxes, wsum, weighted feature sum, Wv matvec.
// ---------------------------------------------------------------------------
__global__ void __launch_bounds__(256)
k_attention(const float* __restrict__ features,
            const int* __restrict__ task_ids,
            const float* __restrict__ Wv,
            const float* __restrict__ bv,
            const float* __restrict__ qk2,
            float* __restrict__ wsum_g,
            float* __restrict__ out_g) {
    // qk_lds padded to 16 rows (rows 4..15 zero) so every lane's B-fragment
    // read is an unconditional ds_load_b64 -> no EXEC save/restore in the
    // WMMA inner loop.
    __shared__ float qk_lds[16][D_];   // 16 KB
    __shared__ float sc[T_][N_];       // 16 KB (scores -> softmax weights)
    __shared__ float red[256];         //  1 KB
    __shared__ float fb_lds[D_];       //  1 KB

    const int b = blockIdx.x;
    const int tid = threadIdx.x;
    const int s = task_ids[b];

    // Stage the 4 qk vectors (column s); zero the 12 padding rows.
    for (int i = tid; i < 16 * D_; i += 256) {
        int t = i >> 8, d = i & (D_ - 1);
        qk_lds[t][d] = (t < T_) ? qk2[((size_t)t * T_ + s) * D_ + d] : 0.f;
    }
    __syncthreads();

    // ---- scores via V_WMMA_F32_16X16X4_F32 -------------------------------
    // A (16x4, MxK): lanes 0-15 hold K=0,1; lanes 16-31 hold K=2,3 (ISA tbl)
    // B (4x16, KxN): n = lane&15 (task index, zeros for n>=4), same K split
    // C/D (16x16): VGPR v -> row m0 + v + 8*half, col = lane&15
    const int lane = tid & 31;
    const int wave = tid >> 5;          // 8 waves: 128 rows per tile pass
    const int half = lane >> 4;
    const int l16  = lane & 15;
    const float* Fb = features + (size_t)b * N_ * D_;
    const float invScale = 0.0625f;     // 1/sqrt(D)

    for (int tile = 0; tile < N_ / 128; ++tile) {
        const int m0 = tile * 128 + wave * 16;
        const float* arow = Fb + (size_t)(m0 + l16) * D_ + half * 2;
        const float* brow = &qk_lds[l16][half * 2];
        v8f acc = {0.f, 0.f, 0.f, 0.f, 0.f, 0.f, 0.f, 0.f};
        for (int k0 = 0; k0 < D_; k0 += 4) {
            v2f a  = *(const v2f*)(arow + k0);
            v2f bb = *(const v2f*)(brow + k0);
            acc = __builtin_amdgcn_wmma_f32_16x16x4_f32(
                false, a, false, bb, (short)0, acc, false, false);
        }
        if (l16 < T_) {
#pragma unroll
            for (int v = 0; v < 8; ++v)
                sc[l16][m0 + half * 8 + v] = acc[v] * invScale;
        }
    }
    __syncthreads();

    // ---- softmax over m for each task t ----------------------------------
    for (int t = 0; t < T_; ++t) {
        float mx = -3.4e38f;
        for (int m = tid; m < N_; m += 256) mx = fmaxf(mx, sc[t][m]);
        red[tid] = mx; __syncthreads();
        for (int off = 128; off > 0; off >>= 1) {
            if (tid < off) red[tid] = fmaxf(red[tid], red[tid + off]);
            __syncthreads();
        }
        mx = red[0]; __syncthreads();

        float sum = 0.f;
        for (int m = tid; m < N_; m += 256) {
            float e = __expf(sc[t][m] - mx);
            sc[t][m] = e;
            sum += e;
        }
        red[tid] = sum; __syncthreads();
        for (int off = 128; off > 0; off >>= 1) {
            if (tid < off) red[tid] += red[tid + off];
            __syncthreads();
        }
        float inv = 1.f / red[0]; __syncthreads();
        for (int m = tid; m < N_; m += 256) sc[t][m] *= inv;
        __syncthreads();
    }

    // ---- wsum over tasks (for final_weights) -----------------------------
    for (int m = tid; m < N_; m += 256)
        wsum_g[(size_t)b * N_ + m] = sc[0][m] + sc[1][m] + sc[2][m] + sc[3][m];

    // ---- fbar[d] = sum_m w[s][m] * F[m,d]  (coalesced over d, hits L2) ---
    float fb = 0.f;
    for (int m = 0; m < N_; ++m)
        fb += sc[s][m] * Fb[(size_t)m * D_ + tid];
    fb_lds[tid] = fb;
    __syncthreads();

    // ---- out[o] = bv[s][o] + dot(Wv[s][o][:], fbar)  (softmax sums to 1) -
    const float* wv = Wv + ((size_t)s * D_ + tid) * D_;
    float ov = bv[s * D_ + tid];
    for (int d = 0; d < D_; d += 4) {
        v4f w = *(const v4f*)(wv + d);
        ov += w.x * fb_lds[d] + w.y * fb_lds[d + 1] +
              w.z * fb_lds[d + 2] + w.w * fb_lds[d + 3];
    }
    out_g[(size_t)b * D_ + tid] = ov;
}

// ---------------------------------------------------------------------------
// K3a: final_features[b,n,:] = out[b,:]   (64 MB NT streaming stores)
// ---------------------------------------------------------------------------
__global__ void __launch_bounds__(256)
k_bcast_features(const float* __restrict__ out_g, float* __restrict__ dst) {
    size_t gid = (size_t)blockIdx.x * 256 + threadIdx.x;  // over B*N*D/4
    int b  = (int)(gid >> 16);   // N*D/4 = 65536
    int d4 = (int)(gid & 63);    // D/4   = 64
    v4f v = *(const v4f*)(out_g + (size_t)b * D_ + d4 * 4);
    __builtin_nontemporal_store(v, (v4f*)dst + gid);
}

// ---------------------------------------------------------------------------
// K3b: final_weights[b,n,:] = wsum[b,:]   (256 MB NT streaming stores)
// ---------------------------------------------------------------------------
__global__ void __launch_bounds__(256)
k_bcast_weights(const float* __restrict__ wsum_g, float* __restrict__ dst) {
    size_t gid = (size_t)blockIdx.x * 256 + threadIdx.x;  // over B*N*N/4
    int b  = (int)(gid >> 18);   // N*N/4 = 262144
    int c4 = (int)(gid & 255);   // N/4   = 256
    v4f v = *(const v4f*)(wsum_g + (size_t)b * N_ + c4 * 4);
    __builtin_nontemporal_store(v, (v4f*)dst + gid);
}

// ---------------------------------------------------------------------------
extern "C" void kernel_launch(void* const* d_in, const int* in_sizes, int n_in,
                              void* d_out, int out_size, void* d_ws, size_t ws_size,
                              hipStream_t stream) {
    (void)in_sizes; (void)n_in; (void)out_size; (void)ws_size;
    const float* features = (const float*)d_in[0];
    const int*   task_ids = (const int*)d_in[1];
    const float* emb      = (const float*)d_in[2];
    const float* Wq       = (const float*)d_in[3];
    const float* bq       = (const float*)d_in[4];
    const float* Wk       = (const float*)d_in[5];
    // d_in[6] (bk) intentionally unused: constant per-row shift cancels in softmax
    const float* Wv       = (const float*)d_in[7];
    const float* bv       = (const float*)d_in[8];

    float* ws   = (float*)d_ws;
    float* qk2  = ws;                      // 16*256 floats
    float* wsum = ws + 16 * D_;            // B*N floats
    float* outv = wsum + (size_t)B_ * N_;  // B*D floats

    float* out_features = (float*)d_out;                        // [B,N,D]
    float* out_weights  = out_features + (size_t)B_ * N_ * D_;  // [B,N,N]

    k_precompute_qk<<<T_ * T_, 256, 0, stream>>>(emb, Wq, bq, Wk, qk2);
    k_attention<<<B_, 256, 0, stream>>>(features, task_ids, Wv, bv, qk2,
                                        wsum, outv);
    k_bcast_features<<<(int)(((size_t)B_ * N_ * D_ / 4) / 256), 256, 0, stream>>>(
        outv, out_features);
    k_bcast_weights<<<(int)(((size_t)B_ * N_ * N_ / 4) / 256), 256, 0, stream>>>(
        wsum, out_weights);
}